// GNN_33921651703886
// MI455X (gfx1250) — compile-verified
//
#include <hip/hip_runtime.h>
#include <hip/hip_bf16.h>

// ---------------------------------------------------------------------------
// 4-layer GCN forward for gfx1250 (MI455X).
//   per layer: h = x @ W[l]           (fp32 WMMA 16x16x4)
//              acc = scatter_add(h[src] * dinv[src]*dinv[dst], dst)
//              x   = relu(acc + h*dinv^2 (self loop) + b[l] + x_prev)
// ---------------------------------------------------------------------------

#define N_NODES 50000
#define EMB_DIM 128
#define N_LAYERS 4

typedef __attribute__((ext_vector_type(2))) float v2f;
typedef __attribute__((ext_vector_type(8))) float v8f;

// ---------------------------------------------------------------------------
// degree / norm preparation
// ---------------------------------------------------------------------------
__global__ __launch_bounds__(256) void gcn_init_deg(float* __restrict__ deg, int n) {
    int i = blockIdx.x * blockDim.x + threadIdx.x;
    if (i < n) deg[i] = 1.0f;   // self-loop contributes 1 to every node's degree
}

__global__ __launch_bounds__(256) void gcn_count_deg(const long long* __restrict__ dst,
                                                     float* __restrict__ deg,
                                                     int nEdges) {
    int e = blockIdx.x * blockDim.x + threadIdx.x;
    if (e < nEdges) {
        int d = (int)dst[e];
        unsafeAtomicAdd(&deg[d], 1.0f);   // native global_atomic_add_f32
    }
}

__global__ __launch_bounds__(256) void gcn_dinv(float* __restrict__ deg, int n) {
    int i = blockIdx.x * blockDim.x + threadIdx.x;
    if (i < n) deg[i] = __frsqrt_rn(deg[i]);   // deg >= 1 always
}

// ---------------------------------------------------------------------------
// h = X @ W    X:[N,128] fp32, W:[128,128] fp32, via V_WMMA_F32_16X16X4_F32.
// One wave per 16x16 output tile; 32 k-steps of K=4.
// N_NODES (50000) is divisible by 16 so the grid is exact -> EXEC all ones.
// ---------------------------------------------------------------------------
__global__ __launch_bounds__(256) void gcn_gemm_wmma(const float* __restrict__ X,
                                                     const float* __restrict__ W,
                                                     float* __restrict__ H) {
    const int tilesN = EMB_DIM / 16;                       // 8
    int wave = (blockIdx.x * blockDim.x + threadIdx.x) >> 5;
    int lane = threadIdx.x & 31;
    int tileM = wave / tilesN;                             // node-tile (uniform per wave)
    int tileN = wave % tilesN;                             // feature-tile

    int mlane  = lane & 15;
    int khalf  = (lane >> 4) << 1;                         // 0 for lanes 0-15, 2 for 16-31
    int row    = tileM * 16 + mlane;                       // A: row M = lane%16 (both halves)
    int ncol   = tileN * 16 + mlane;                       // B: col N = lane%16 (both halves)
    const float* xrow = X + (long long)row * EMB_DIM;

    v8f c = {};
    #pragma unroll 4
    for (int k0 = 0; k0 < EMB_DIM; k0 += 4) {
        // A fragment: 16x4 fp32; lane holds x[row, k0+khalf .. k0+khalf+1]
        v2f a;
        a.x = xrow[k0 + khalf + 0];
        a.y = xrow[k0 + khalf + 1];
        // B fragment: 4x16 fp32; lane holds W[k0+khalf .. +1, ncol]
        v2f bb;
        bb.x = W[(k0 + khalf + 0) * EMB_DIM + ncol];
        bb.y = W[(k0 + khalf + 1) * EMB_DIM + ncol];
        // (neg_a, A, neg_b, B, c_mod, C, reuse_a, reuse_b)
        c = __builtin_amdgcn_wmma_f32_16x16x4_f32(false, a, false, bb,
                                                  (short)0, c, false, false);
    }

    // C/D layout: VGPR r -> (M = r + 8*(lane>=16), N = lane%16)
    int mbase = tileM * 16 + ((lane >> 4) << 3);
    #pragma unroll
    for (int r = 0; r < 8; ++r) {
        H[(long long)(mbase + r) * EMB_DIM + (tileN * 16 + mlane)] = c[r];
    }
}

// ---------------------------------------------------------------------------
// zero the accumulator
// ---------------------------------------------------------------------------
__global__ __launch_bounds__(256) void gcn_zero(float* __restrict__ p, long long n) {
    long long i = (long long)blockIdx.x * blockDim.x + threadIdx.x;
    if (i < n) p[i] = 0.0f;
}

// ---------------------------------------------------------------------------
// Edge scatter: one wave per edge, each lane owns a float4 slice of the
// 128-dim feature row (32 lanes * 4 = 128). h and acc are L2-resident.
// ---------------------------------------------------------------------------
__global__ __launch_bounds__(256) void gcn_scatter(const long long* __restrict__ src,
                                                   const long long* __restrict__ dst,
                                                   const float* __restrict__ dinv,
                                                   const float* __restrict__ H,
                                                   float* __restrict__ ACC,
                                                   int nEdges) {
    int wave = (blockIdx.x * blockDim.x + threadIdx.x) >> 5;
    int lane = threadIdx.x & 31;
    if (wave >= nEdges) return;

    int s = (int)src[wave];
    int d = (int)dst[wave];
    float norm = dinv[s] * dinv[d];

    const float4* hs = (const float4*)(H + (long long)s * EMB_DIM);
    float4 v = hs[lane];                       // coalesced 128B/wave *4
    float* a = ACC + (long long)d * EMB_DIM + lane * 4;
    unsafeAtomicAdd(a + 0, v.x * norm);
    unsafeAtomicAdd(a + 1, v.y * norm);
    unsafeAtomicAdd(a + 2, v.z * norm);
    unsafeAtomicAdd(a + 3, v.w * norm);
}

// ---------------------------------------------------------------------------
// combine: out = relu(acc + h*dinv^2 (self loop) + bias + prev)
// ---------------------------------------------------------------------------
__global__ __launch_bounds__(256) void gcn_combine(const float* __restrict__ ACC,
                                                   const float* __restrict__ H,
                                                   const float* __restrict__ dinv,
                                                   const float* __restrict__ bias,
                                                   const float* __restrict__ xprev,
                                                   float* __restrict__ xout,
                                                   long long nTot) {
    long long i = (long long)blockIdx.x * blockDim.x + threadIdx.x;
    if (i >= nTot) return;
    int f = (int)(i & (EMB_DIM - 1));
    int n = (int)(i >> 7);
    float di = dinv[n];
    float v = ACC[i] + H[i] * di * di + bias[f] + xprev[i];
    xout[i] = fmaxf(v, 0.0f);
}

// ---------------------------------------------------------------------------
// host launcher
// ---------------------------------------------------------------------------
extern "C" void kernel_launch(void* const* d_in, const int* in_sizes, int n_in,
                              void* d_out, int out_size, void* d_ws, size_t ws_size,
                              hipStream_t stream) {
    const float*     x    = (const float*)d_in[0];
    const long long* eidx = (const long long*)d_in[1];   // int64 [2, E]
    const float*     W    = (const float*)d_in[2];       // [L, 128, 128]
    const float*     bias = (const float*)d_in[3];       // [L, 128]

    const int  N = N_NODES;
    const int  E = in_sizes[1] / 2;
    const long long NT = (long long)N * EMB_DIM;

    const long long* srcIdx = eidx;       // edge_index[0]
    const long long* dstIdx = eidx + E;   // edge_index[1]

    // workspace layout (256B aligned slabs)
    char* ws = (char*)d_ws;
    size_t off = 0;
    float* dinv = (float*)(ws + off); off += ((size_t)N * 4 + 255) & ~(size_t)255;
    float* h    = (float*)(ws + off); off += ((size_t)NT * 4 + 255) & ~(size_t)255;
    float* acc  = (float*)(ws + off); off += ((size_t)NT * 4 + 255) & ~(size_t)255;
    float* xcur = (float*)(ws + off);

    // ---- degrees / normalization (once per call, deterministic) ----
    gcn_init_deg<<<(N + 255) / 256, 256, 0, stream>>>(dinv, N);
    gcn_count_deg<<<(E + 255) / 256, 256, 0, stream>>>(dstIdx, dinv, E);
    gcn_dinv<<<(N + 255) / 256, 256, 0, stream>>>(dinv, N);

    const int gemmBlocks    = (N / 16) * (EMB_DIM / 16) * 32 / 256;  // 3125
    const int zeroBlocks    = (int)((NT + 255) / 256);
    const int scatterBlocks = (E * 32 + 255) / 256;                  // one wave per edge
    const int combBlocks    = (int)((NT + 255) / 256);

    for (int l = 0; l < N_LAYERS; ++l) {
        const float* xin  = (l == 0) ? x : xcur;
        float*       xout = (l == N_LAYERS - 1) ? (float*)d_out : xcur;
        const float* Wl   = W + (size_t)l * EMB_DIM * EMB_DIM;
        const float* bl   = bias + (size_t)l * EMB_DIM;

        gcn_gemm_wmma<<<gemmBlocks, 256, 0, stream>>>(xin, Wl, h);
        gcn_zero<<<zeroBlocks, 256, 0, stream>>>(acc, NT);
        gcn_scatter<<<scatterBlocks, 256, 0, stream>>>(srcIdx, dstIdx, dinv, h, acc, E);
        gcn_combine<<<combBlocks, 256, 0, stream>>>(acc, h, dinv, bl, xin, xout, NT);
    }
}